// TransE_MC_32126355374174
// MI455X (gfx1250) — compile-verified
//
#include <hip/hip_runtime.h>
#include <cstdint>
#include <cstddef>

#define NUM_ENT 50000
#define RANK    128
#define BATCH   64

#define TE        64                 // entities per LDS tile
#define EROW      132                // padded row stride (floats): 16B-aligned rows, banks spread
#define QFLOATS   (BATCH * RANK)     // 8192 floats = 32KB
#define TILE_FL   (TE * EROW)        // 8448 floats = 33792B
#define NTILES    ((NUM_ENT + TE - 1) / TE)   // 782
#define THREADS   256
#define GRIDX     196

// Low 32 bits of a generic pointer to LDS == LDS byte offset (hardware adds wave LDS_BASE).
__device__ __forceinline__ uint32_t lds_off(const void* p) {
  return (uint32_t)(uintptr_t)p;
}

// CDNA5 async copy global -> LDS, tracked by ASYNCcnt (CDNA5 ISA).
// vdst operand = per-lane LDS byte offset, vaddr = per-lane 64-bit global address.
__device__ __forceinline__ void async_ld_b128(uint32_t lds_byte_off, const void* gaddr) {
  asm volatile("global_load_async_to_lds_b128 %0, %1, off"
               :: "v"(lds_byte_off), "v"(gaddr) : "memory");
}

// Issue one 64x128 f32 entity tile (32KB payload) as 8 b128 async loads per thread.
__device__ __forceinline__ void issue_tile(uint32_t lds_base, const float* __restrict__ Eg,
                                           int tile, int tid) {
#pragma unroll
  for (int r = 0; r < 8; ++r) {
    int c   = tid + r * THREADS;     // chunk 0..2047 (float4 granules)
    int e   = c >> 5;                // row in tile, 0..63
    int k4  = (c & 31) << 2;         // k start, 0..124
    int eg  = tile * TE + e;
    if (eg >= NUM_ENT) eg = 0;       // clamp tail rows; never stored
    const float* gp = Eg + (size_t)eg * RANK + k4;
    uint32_t lo = lds_base + (uint32_t)(e * EROW + k4) * 4u;
    async_ld_b128(lo, gp);
  }
}

__global__ __launch_bounds__(THREADS)
void transe_l1_scores_kernel(const int*   __restrict__ x,
                             const float* __restrict__ lhs_w,
                             const float* __restrict__ rel_w,
                             const float* __restrict__ rhs_w,
                             float*       __restrict__ out) {
  extern __shared__ float smem[];
  float* Q    = smem;                     // [64][128] query block
  float* buf0 = smem + QFLOATS;           // entity tile, double-buffered
  float* buf1 = buf0 + TILE_FL;

  const int sel = blockIdx.y;             // 0: scores_sp vs rhs_w, 1: scores_po vs lhs_w
  const float* E = (sel == 0) ? rhs_w : lhs_w;
  float* outp = out + (size_t)sel * (size_t)BATCH * (size_t)NUM_ENT;
  const int tid = threadIdx.x;

  // Build query block in LDS (float4-granular, compile-time trip count):
  //  sel 0: q = lhs_w[x0] + rel_w[x1]        (sp: -sum|q - rhs_w[e]|)
  //  sel 1: q = rhs_w[x2] - rel_w[x1]        (po: -sum|lhs_w[e] - q| == -sum|lhs_w[e] + rel - rhs|)
#pragma unroll
  for (int r = 0; r < QFLOATS / 4 / THREADS; ++r) {       // 8 iterations
    int c  = tid + r * THREADS;                           // float4 chunk 0..2047
    int b  = c >> 5;                                      // 32 float4 per row
    int k4 = (c & 31) << 2;
    int i0 = x[b * 3 + 0];
    int i1 = x[b * 3 + 1];
    int i2 = x[b * 3 + 2];
    float4 rv = *(const float4*)(rel_w + (size_t)i1 * RANK + k4);
    float4 v;
    if (sel == 0) {
      float4 lv = *(const float4*)(lhs_w + (size_t)i0 * RANK + k4);
      v = make_float4(lv.x + rv.x, lv.y + rv.y, lv.z + rv.z, lv.w + rv.w);
    } else {
      float4 hv = *(const float4*)(rhs_w + (size_t)i2 * RANK + k4);
      v = make_float4(hv.x - rv.x, hv.y - rv.y, hv.z - rv.z, hv.w - rv.w);
    }
    *(float4*)(Q + (size_t)b * RANK + k4) = v;            // ds_store_b128
  }

  const uint32_t boff[2] = { lds_off(buf0), lds_off(buf1) };
  const int em = tid & 15;                 // entity lane within tile (rows em + 16j)
  const int bg = (tid >> 4) << 2;          // batch base (4 rows per thread)
  const float4* Qv = (const float4*)Q;

  int bufid = 0;
  issue_tile(boff[0], E, blockIdx.x, tid); // prologue prefetch

  for (int t = blockIdx.x; t < NTILES; t += gridDim.x) {
    const int tn = t + gridDim.x;
    if (tn < NTILES) {
      issue_tile(boff[bufid ^ 1], E, tn, tid);
      asm volatile("s_wait_asynccnt 8" ::: "memory");   // current tile's 8 loads done
    } else {
      asm volatile("s_wait_asynccnt 0" ::: "memory");
    }
    __syncthreads();                                    // tile (and Q on iter 0) visible

    const float* Eb = (bufid == 0) ? buf0 : buf1;
    float acc[4][4];
#pragma unroll
    for (int i = 0; i < 4; ++i)
#pragma unroll
      for (int j = 0; j < 4; ++j) acc[i][j] = 0.0f;

#pragma unroll 8
    for (int k4 = 0; k4 < RANK / 4; ++k4) {
      float4 q[4], ev[4];
#pragma unroll
      for (int i = 0; i < 4; ++i)
        q[i] = Qv[(bg + i) * (RANK / 4) + k4];          // broadcast across half-wave
#pragma unroll
      for (int j = 0; j < 4; ++j)                       // rows em+16j: 16 distinct bank-quads
        ev[j] = *(const float4*)(Eb + (em + 16 * j) * EROW + (k4 << 2));
#pragma unroll
      for (int i = 0; i < 4; ++i)
#pragma unroll
        for (int j = 0; j < 4; ++j) {
          acc[i][j] += __builtin_fabsf(q[i].x - ev[j].x);
          acc[i][j] += __builtin_fabsf(q[i].y - ev[j].y);
          acc[i][j] += __builtin_fabsf(q[i].z - ev[j].z);
          acc[i][j] += __builtin_fabsf(q[i].w - ev[j].w);
        }
    }

    const int ebase = t * TE + em;
#pragma unroll
    for (int j = 0; j < 4; ++j) {
      const int eg = ebase + 16 * j;
      if (eg < NUM_ENT) {
#pragma unroll
        for (int i = 0; i < 4; ++i)
          __builtin_nontemporal_store(-acc[i][j],
                                      outp + (size_t)(bg + i) * NUM_ENT + eg);
      }
    }

    __syncthreads();           // done reading this buffer before it is refilled
    bufid ^= 1;
  }
}

extern "C" void kernel_launch(void* const* d_in, const int* in_sizes, int n_in,
                              void* d_out, int out_size, void* d_ws, size_t ws_size,
                              hipStream_t stream) {
  (void)in_sizes; (void)n_in; (void)out_size; (void)d_ws; (void)ws_size;
  const int*   x     = (const int*)d_in[0];
  const float* lhs_w = (const float*)d_in[1];
  const float* rel_w = (const float*)d_in[2];
  const float* rhs_w = (const float*)d_in[3];
  float* out = (float*)d_out;

  const size_t shmem = (size_t)(QFLOATS + 2 * TILE_FL) * sizeof(float);  // 100352 B < 320KB/WGP
  dim3 grid(GRIDX, 2);
  transe_l1_scores_kernel<<<grid, THREADS, shmem, stream>>>(x, lhs_w, rel_w, rhs_w, out);
}